// VariationalSAGEEncoder_54597624267031
// MI455X (gfx1250) — compile-verified
//
#include <hip/hip_runtime.h>

typedef __attribute__((ext_vector_type(2))) float v2f;
typedef __attribute__((ext_vector_type(8))) float v8f;

#define N_NODES 100000
#define IN_CH   128
#define HID     128
#define OUT_CH  64
#define N_EDGES 1000000
#define LDSS    132   // padded LDS row stride: 132 % 64 == 4 -> conflict-free A-frag reads

// ---------------------------------------------------------------- degree count
__global__ void deg_kernel(const long long* __restrict__ dst, float* __restrict__ deg) {
  int e = blockIdx.x * blockDim.x + threadIdx.x;
  if (e < N_EDGES) atomicAdd(&deg[(int)dst[e]], 1.0f);
}

__global__ void invdeg_kernel(const float* __restrict__ deg, float* __restrict__ invdeg) {
  int i = blockIdx.x * blockDim.x + threadIdx.x;
  if (i < N_NODES) invdeg[i] = 1.0f / fmaxf(deg[i], 1.0f);
}

// ------------------------------------------------- scatter-add 128-wide features
// thread -> (edge, feature); consecutive lanes cover one feature row (coalesced
// gather of feat[src] and coalesced f32 atomics into agg[dst]).
__global__ void scatter_kernel(const float* __restrict__ feat,
                               const long long* __restrict__ src,
                               const long long* __restrict__ dst,
                               float* __restrict__ agg) {
  long long gid = (long long)blockIdx.x * blockDim.x + threadIdx.x;
  int e = (int)(gid >> 7);
  int f = (int)(gid & 127);
  if (e < N_EDGES) {
    int s = (int)src[e];
    int d = (int)dst[e];
    atomicAdd(&agg[(long long)d * IN_CH + f], feat[(long long)s * IN_CH + f]);
  }
}

// ---------------------------------------------------------------- fused GEMM 1
// h = relu( (agg*invdeg) @ Wl + x @ Wr + b ),  [N_NODES x 128] x [128 x 128]
// block = 8 waves; block owns one 16-row M-tile, wave w owns N-tile w (16 cols).
__global__ __launch_bounds__(256)
void sage_gemm1(const float* __restrict__ agg, const float* __restrict__ invdeg,
                const float* __restrict__ x,
                const float* __restrict__ Wl, const float* __restrict__ Wr,
                const float* __restrict__ b, float* __restrict__ h) {
  __shared__ float sAgg[16 * LDSS];
  __shared__ float sX[16 * LDSS];
  const int mbase = blockIdx.x * 16;
  const int tid = threadIdx.x;

#pragma unroll
  for (int i = 0; i < 8; ++i) {           // 2048 floats per matrix, 8 per thread
    int idx = tid + i * 256;
    int row = idx >> 7, col = idx & 127;
    float inv = invdeg[mbase + row];      // fuse mean-scaling into the stage
    sAgg[row * LDSS + col] = agg[(long long)(mbase + row) * HID + col] * inv;
    sX[row * LDSS + col]   = x[(long long)(mbase + row) * HID + col];
  }
  __syncthreads();

  const int lane  = tid & 31;
  const int wave  = tid >> 5;             // 0..7 -> N-tile
  const int m     = lane & 15;            // A-frag row
  const int khalf = (lane >> 4) << 1;     // lanes 16-31 hold K+2,K+3
  const int n     = (wave << 4) + (lane & 15);

  v8f c = {};
#pragma unroll 4
  for (int k0 = 0; k0 < HID; k0 += 4) {
    const int kk = k0 + khalf;
    v2f aA, aX, bL, bR;
    aA.x = sAgg[m * LDSS + kk]; aA.y = sAgg[m * LDSS + kk + 1];
    aX.x = sX[m * LDSS + kk];   aX.y = sX[m * LDSS + kk + 1];
    bL.x = Wl[kk * HID + n];    bL.y = Wl[(kk + 1) * HID + n];
    bR.x = Wr[kk * HID + n];    bR.y = Wr[(kk + 1) * HID + n];
    c = __builtin_amdgcn_wmma_f32_16x16x4_f32(false, aA, false, bL, (short)0, c, false, false);
    c = __builtin_amdgcn_wmma_f32_16x16x4_f32(false, aX, false, bR, (short)0, c, false, false);
  }

  const float bias = b[n];
#pragma unroll
  for (int r = 0; r < 8; ++r) {
    int mo = mbase + r + ((lane >> 4) << 3);   // C/D layout: M = r + 8*(lane>=16)
    float v = c[r] + bias;
    h[(long long)mo * HID + n] = v > 0.0f ? v : 0.0f;
  }
}

// ---------------------------------------------------------------- fused GEMM 2
// mu     = (agg_h*invdeg) @ Wmu_l + h @ Wmu_r + bmu   [N x 64]
// logstd = (agg_h*invdeg) @ Wls_l + h @ Wls_r + bls   [N x 64]
// block = 8 waves on one 16-row M-tile; waves 0-3 -> mu N-tiles, 4-7 -> logstd.
__global__ __launch_bounds__(256)
void sage_gemm2(const float* __restrict__ agg, const float* __restrict__ invdeg,
                const float* __restrict__ h,
                const float* __restrict__ Wmu_l, const float* __restrict__ Wmu_r,
                const float* __restrict__ bmu,
                const float* __restrict__ Wls_l, const float* __restrict__ Wls_r,
                const float* __restrict__ bls,
                float* __restrict__ mu, float* __restrict__ ls) {
  __shared__ float sAgg[16 * LDSS];
  __shared__ float sH[16 * LDSS];
  const int mbase = blockIdx.x * 16;
  const int tid = threadIdx.x;

#pragma unroll
  for (int i = 0; i < 8; ++i) {
    int idx = tid + i * 256;
    int row = idx >> 7, col = idx & 127;
    float inv = invdeg[mbase + row];
    sAgg[row * LDSS + col] = agg[(long long)(mbase + row) * HID + col] * inv;
    sH[row * LDSS + col]   = h[(long long)(mbase + row) * HID + col];
  }
  __syncthreads();

  const int lane  = tid & 31;
  const int wave  = tid >> 5;
  const int half  = wave >> 2;            // 0 = mu, 1 = logstd
  const int nt    = wave & 3;
  const int m     = lane & 15;
  const int khalf = (lane >> 4) << 1;
  const int n     = (nt << 4) + (lane & 15);

  const float* __restrict__ Wl  = half ? Wls_l : Wmu_l;
  const float* __restrict__ Wr  = half ? Wls_r : Wmu_r;
  const float* __restrict__ bb  = half ? bls   : bmu;
  float* __restrict__ out       = half ? ls    : mu;

  v8f c = {};
#pragma unroll 4
  for (int k0 = 0; k0 < HID; k0 += 4) {
    const int kk = k0 + khalf;
    v2f aA, aH, bL, bR;
    aA.x = sAgg[m * LDSS + kk]; aA.y = sAgg[m * LDSS + kk + 1];
    aH.x = sH[m * LDSS + kk];   aH.y = sH[m * LDSS + kk + 1];
    bL.x = Wl[kk * OUT_CH + n]; bL.y = Wl[(kk + 1) * OUT_CH + n];
    bR.x = Wr[kk * OUT_CH + n]; bR.y = Wr[(kk + 1) * OUT_CH + n];
    c = __builtin_amdgcn_wmma_f32_16x16x4_f32(false, aA, false, bL, (short)0, c, false, false);
    c = __builtin_amdgcn_wmma_f32_16x16x4_f32(false, aH, false, bR, (short)0, c, false, false);
  }

  const float bias = bb[n];
#pragma unroll
  for (int r = 0; r < 8; ++r) {
    int mo = mbase + r + ((lane >> 4) << 3);
    out[(long long)mo * OUT_CH + n] = c[r] + bias;
  }
}

// ------------------------------------------------------------------- launcher
extern "C" void kernel_launch(void* const* d_in, const int* in_sizes, int n_in,
                              void* d_out, int out_size, void* d_ws, size_t ws_size,
                              hipStream_t stream) {
  const float*     x    = (const float*)d_in[0];
  const long long* ei   = (const long long*)d_in[1];   // int64 [2, N_EDGES]
  const float*     W1l  = (const float*)d_in[2];
  const float*     W1r  = (const float*)d_in[3];
  const float*     b1   = (const float*)d_in[4];
  const float*     Wmul = (const float*)d_in[5];
  const float*     Wmur = (const float*)d_in[6];
  const float*     bmu  = (const float*)d_in[7];
  const float*     Wlsl = (const float*)d_in[8];
  const float*     Wlsr = (const float*)d_in[9];
  const float*     bls  = (const float*)d_in[10];

  const long long* src = ei;
  const long long* dst = ei + N_EDGES;

  // workspace carve-out (agg buffer is reused for layer 1 and layer 2)
  char* ws = (char*)d_ws;
  size_t off = 0;
  float* deg    = (float*)(ws + off); off += (size_t)N_NODES * sizeof(float);
  off = (off + 255) & ~(size_t)255;
  float* invdeg = (float*)(ws + off); off += (size_t)N_NODES * sizeof(float);
  off = (off + 255) & ~(size_t)255;
  float* agg    = (float*)(ws + off); off += (size_t)N_NODES * HID * sizeof(float);
  off = (off + 255) & ~(size_t)255;
  float* h      = (float*)(ws + off); off += (size_t)N_NODES * HID * sizeof(float);

  float* mu = (float*)d_out;                         // [N_NODES x 64]
  float* ls = (float*)d_out + (size_t)N_NODES * OUT_CH;

  const unsigned scatterGrid = (unsigned)(((long long)N_EDGES * 128 + 255) / 256);

  // ---- layer 1: aggregate x ------------------------------------------------
  hipMemsetAsync(deg, 0, (size_t)N_NODES * sizeof(float), stream);
  hipMemsetAsync(agg, 0, (size_t)N_NODES * HID * sizeof(float), stream);
  deg_kernel<<<(N_EDGES + 255) / 256, 256, 0, stream>>>(dst, deg);
  scatter_kernel<<<scatterGrid, 256, 0, stream>>>(x, src, dst, agg);
  invdeg_kernel<<<(N_NODES + 255) / 256, 256, 0, stream>>>(deg, invdeg);
  sage_gemm1<<<N_NODES / 16, 256, 0, stream>>>(agg, invdeg, x, W1l, W1r, b1, h);

  // ---- layer 2: aggregate h (shared by mu and logstd) ----------------------
  hipMemsetAsync(agg, 0, (size_t)N_NODES * HID * sizeof(float), stream);
  scatter_kernel<<<scatterGrid, 256, 0, stream>>>(h, src, dst, agg);
  sage_gemm2<<<N_NODES / 16, 256, 0, stream>>>(agg, invdeg, h,
                                               Wmul, Wmur, bmu,
                                               Wlsl, Wlsr, bls,
                                               mu, ls);
}